// HstackDiagProduct_21225728377283
// MI455X (gfx1250) — compile-verified
//
#include <hip/hip_runtime.h>

// ---------------------------------------------------------------------------
// HstackDiagProduct on MI455X (gfx1250, wave32).
//
// Fully fused: one workgroup (256 thr = 8 waves) owns 8 batch rows; all 10
// butterfly stages run with state ping-ponged in LDS (2x128KB + 36KB kernels
// ~= 293KB of 320KB/WGP).
//  * Stages 0..4 (3% FLOPs, VALU-bound): register-tiled polynomial products,
//    one full output entry per thread, fully unrolled (1 v_fmac per MAC),
//    kernels staged compactly in LDS.
//  * Stages 5..9 (97% FLOPs): fp32 WMMA 16x16x4 GEMMs vs Toeplitz fragments
//    from REVERSED zero-padded kernels in LDS (both B taps ascending in K ->
//    ds_load_2addr lands in operand order). State carries an r*4 skew so all
//    32 lanes of the A ds_load_b64 hit distinct banks. K-loop SGPR-uniform
//    (EXEC all-ones around v_wmma), band-restricted, 4 independent
//    accumulator chains (src,m) for ILP.
// ---------------------------------------------------------------------------

#define B_TILE   8
#define BSTRIDE  4104   // 4096 + 8 floats of slack (r-skew + bank stagger)
#define NTHREADS 256

typedef __attribute__((ext_vector_type(2))) float v2f;
typedef __attribute__((ext_vector_type(4))) float v4f;
typedef __attribute__((ext_vector_type(8))) float v8f;

struct PolyArgs {
    const float* in;
    const float* d1[10];
    const float* d2[10];
    const float* P;
    float*       out;
};

// ---- stages 0..4: register-tiled polynomial multiply, fully unrolled ----
template<int J>
__device__ __forceinline__ void valu_stage(const float* __restrict__ d1,
                                           const float* __restrict__ d2,
                                           const float* cur, float* nxt,
                                           float* sW, int tid) {
    constexpr int L    = 1 << J;
    constexpr int twoL = L << 1;
    constexpr int n    = 1024 >> (J + 1);
    constexpr int logn = 9 - J;

    // stage compact kernels: slot = ((i*2+c)*2+src)*2+m, length L+1  (<=32KB)
    constexpr int tot = 8 * n * (L + 1);
    for (int k = tid; k < tot; k += NTHREADS) {
        int slot = k / (L + 1);
        int u    = k - slot * (L + 1);
        int m    = slot & 1;
        int src  = (slot >> 1) & 1;
        int c    = (slot >> 2) & 1;
        int i    = slot >> 3;
        const float* dp = src ? d2 : d1;
        sW[k] = dp[((i * 2 + m) * 2 + c) * (L + 1) + u];
    }
    __syncthreads();

    constexpr int entries = B_TILE * n * 4;   // 16384 >> J, multiple of 256
#pragma unroll 1
    for (int e = tid; e < entries; e += NTHREADS) {
        int c = e & 1, r = (e >> 1) & 1;
        int i = (e >> 2) & (n - 1);
        int b = e >> (2 + logn);
        float acc[twoL];
#pragma unroll
        for (int t = 0; t < twoL; ++t) acc[t] = 0.f;
#pragma unroll
        for (int src = 0; src < 2; ++src) {
            int ib = i + src * n;
#pragma unroll
            for (int m = 0; m < 2; ++m) {
                const float* x = cur + b * BSTRIDE + ((ib * 2 + r) * 2 + m) * L + r * 4;
                const float* w = sW + (((i * 2 + c) * 2 + src) * 2 + m) * (L + 1);
                float xr[L], wr[L + 1];
#pragma unroll
                for (int s = 0; s < L; ++s) xr[s] = x[s];
#pragma unroll
                for (int u = 0; u <= L; ++u) wr[u] = w[u];
#pragma unroll
                for (int s = 0; s < L; ++s)
#pragma unroll
                    for (int u = 0; u <= L; ++u)
                        acc[s + u] += xr[s] * wr[u];
            }
        }
        float* o = nxt + b * BSTRIDE + ((i * 2 + r) * 2 + c) * twoL + r * 4;
#pragma unroll
        for (int t = 0; t < twoL; ++t) o[t] = acc[t];
    }
    __syncthreads();
}

__global__ __launch_bounds__(NTHREADS, 1)
void hstack_diag_product_kernel(PolyArgs A) {
    // ping-pong state: S[b][block i][r][c][t] at offset e*L + r*4 (skew)
    __shared__ float sS[2 * B_TILE * BSTRIDE];   // 256.5 KB
    __shared__ float sW[9216];                   // staged kernels, 36 KB

    const int tid  = threadIdx.x;
    const int lane = tid & 31;
    // wave index as a compiler-known SGPR: keeps all WMMA loop bounds uniform
    const int wv   = __builtin_amdgcn_readfirstlane(tid >> 5);
    const int b0   = blockIdx.x * B_TILE;

    float* cur = sS;
    float* nxt = sS + B_TILE * BSTRIDE;

    // ---- load input tile: b128 global loads, r-skewed LDS stores ----
    for (int k = tid; k < B_TILE * 1024; k += NTHREADS) {
        int b = k >> 10, blk = k & 1023;
        v4f v = *(const v4f*)(A.in + (size_t)(b0 + b) * 4096 + blk * 4);
        float* p = cur + b * BSTRIDE + blk * 4;
        *(v2f*)(p)     = v2f{v.x, v.y};   // r=0 entries (c=0,1)
        *(v2f*)(p + 6) = v2f{v.z, v.w};   // r=1 entries, skew +4
    }
    // warm L2/L0 with all stage weights while the copy drains
    for (int j = 0; j < 10; ++j) {
        const int bytes = (512 >> j) * 16 * ((1 << j) + 1);
        for (int off = tid * 256; off < bytes; off += NTHREADS * 256) {
            __builtin_prefetch((const char*)A.d1[j] + off, 0, 2);
            __builtin_prefetch((const char*)A.d2[j] + off, 0, 2);
        }
    }
    __syncthreads();

    // ---- stages 0..4 ----
    valu_stage<0>(A.d1[0], A.d2[0], cur, nxt, sW, tid); { float* t_ = cur; cur = nxt; nxt = t_; }
    valu_stage<1>(A.d1[1], A.d2[1], cur, nxt, sW, tid); { float* t_ = cur; cur = nxt; nxt = t_; }
    valu_stage<2>(A.d1[2], A.d2[2], cur, nxt, sW, tid); { float* t_ = cur; cur = nxt; nxt = t_; }
    valu_stage<3>(A.d1[3], A.d2[3], cur, nxt, sW, tid); { float* t_ = cur; cur = nxt; nxt = t_; }
    valu_stage<4>(A.d1[4], A.d2[4], cur, nxt, sW, tid); { float* t_ = cur; cur = nxt; nxt = t_; }

    // ---- stages 5..9: fp32 WMMA 16x16x4 vs banded Toeplitz fragments ----
    const int mrow  = lane & 15;          // A: M row / B,C: N col
    const int koff  = (lane >> 4) << 1;   // K sub-offset per half-wave (0 or 2)
    const int bloc  = mrow >> 1;          // local batch row of this lane (A)
    const int rrow  = mrow & 1;           // r index of this lane (A)
    const int mbase = (lane >> 4) << 3;   // C: M base (0 or 8)

    for (int j = 5; j < 10; ++j) {
        const int L = 1 << j;
        const int twoL = L << 1;
        const int n = 1024 >> (j + 1);
        const int KST = L + 40;           // reversed kernel slot w/ both pads
        const int REV0 = L + 19;          // rv[p] = w[REV0 - p]

        // stage the 8n conv kernels into LDS, reversed + zero-padded
        const int tot = 8 * n * KST;
        for (int k = tid; k < tot; k += NTHREADS) {
            int slot = k / KST;
            int pos  = k - slot * KST;
            int m   = slot & 1;
            int src = (slot >> 1) & 1;
            int c   = (slot >> 2) & 1;
            int i   = slot >> 3;
            int u = REV0 - pos;
            float v = 0.f;
            if (u >= 0 && u <= L) {
                const float* dp = src ? A.d2[j] : A.d1[j];
                v = dp[((i * 2 + m) * 2 + c) * (L + 1) + u];
            }
            sW[k] = v;
        }
        __syncthreads();

        // 128 output tiles/stage (n blocks * 2 cols * 2L/16 N-tiles); a wave
        // owns complete K-chains for its tiles (no cross-wave accumulation).
        for (int tile = wv; tile < 128; tile += 8) {
            const int i   = tile >> (j - 2);              // uniform (SGPR)
            const int rem = tile & ((L >> 2) - 1);
            const int c   = rem >> (j - 3);
            const int t0  = (rem & ((L >> 3) - 1)) << 4;

            // nonzero Toeplitz band: s in [max(0,t0-L), min(L-1,t0+15)]
            const int sb0  = (t0 >= L) ? ((t0 - L) & ~3) : 0;
            const int smax = (L - 1 < t0 + 15) ? (L - 1) : (t0 + 15);
            const int sb1  = smax & ~3;

            // A-operand rows for the 4 K-segments (src,m), r-skewed
            const float* x00 = cur + bloc * BSTRIDE + ((i * 2 + rrow) * 2) * L
                             + rrow * 4 + koff;
            const float* x01 = x00 + L;
            const float* x10 = cur + bloc * BSTRIDE + (((i + n) * 2 + rrow) * 2) * L
                             + rrow * 4 + koff;
            const float* x11 = x10 + L;
            // reversed kernels: B[k][N] = rv[REV0 - t0 - N + sb + k]
            const float* w00 = sW + ((i * 2 + c) * 4) * KST + REV0 - t0 - mrow + koff;
            const float* w01 = w00 + KST;
            const float* w10 = w00 + 2 * KST;
            const float* w11 = w00 + 3 * KST;

            v8f acc0 = {0.f,0.f,0.f,0.f,0.f,0.f,0.f,0.f};
            v8f acc1 = acc0, acc2 = acc0, acc3 = acc0;

            // software-pipelined uniform loop (one-ahead prefetch; the
            // one-past-the-end reads stay inside the padded kernel slot / sS)
            v2f a0 = *(const v2f*)(x00 + sb0);
            v2f a1 = *(const v2f*)(x01 + sb0);
            v2f a2 = *(const v2f*)(x10 + sb0);
            v2f a3 = *(const v2f*)(x11 + sb0);
            v2f f0 = {w00[sb0], w00[sb0 + 1]};
            v2f f1 = {w01[sb0], w01[sb0 + 1]};
            v2f f2 = {w10[sb0], w10[sb0 + 1]};
            v2f f3 = {w11[sb0], w11[sb0 + 1]};

            for (int sb = sb0; sb <= sb1; sb += 4) {
                const int sn = sb + 4;
                v2f na0 = *(const v2f*)(x00 + sn);
                v2f na1 = *(const v2f*)(x01 + sn);
                v2f na2 = *(const v2f*)(x10 + sn);
                v2f na3 = *(const v2f*)(x11 + sn);
                v2f nf0 = {w00[sn], w00[sn + 1]};
                v2f nf1 = {w01[sn], w01[sn + 1]};
                v2f nf2 = {w10[sn], w10[sn + 1]};
                v2f nf3 = {w11[sn], w11[sn + 1]};
                acc0 = __builtin_amdgcn_wmma_f32_16x16x4_f32(
                           false, a0, false, f0, (short)0, acc0, false, false);
                acc1 = __builtin_amdgcn_wmma_f32_16x16x4_f32(
                           false, a1, false, f1, (short)0, acc1, false, false);
                acc2 = __builtin_amdgcn_wmma_f32_16x16x4_f32(
                           false, a2, false, f2, (short)0, acc2, false, false);
                acc3 = __builtin_amdgcn_wmma_f32_16x16x4_f32(
                           false, a3, false, f3, (short)0, acc3, false, false);
                a0 = na0; a1 = na1; a2 = na2; a3 = na3;
                f0 = nf0; f1 = nf1; f2 = nf2; f3 = nf3;
            }

            v8f acc = (acc0 + acc1) + (acc2 + acc3);
            for (int v = 0; v < 8; ++v) {
                int Mr = mbase + v;
                int ob = Mr >> 1, orr = Mr & 1;
                nxt[ob * BSTRIDE + ((i * 2 + orr) * 2 + c) * twoL + orr * 4
                    + t0 + mrow] = acc[v];
            }
        }
        __syncthreads();
        float* tmp = cur; cur = nxt; nxt = tmp;
    }

    // ---- epilogue: row 1, drop top coefficient, multiply by P_init ----
    const float P00 = A.P[0], P01 = A.P[1], P10 = A.P[2], P11 = A.P[3];
    for (int k = tid; k < B_TILE * 1024; k += NTHREADS) {
        int b = k >> 10, t = k & 1023;
        const float* x0 = cur + b * BSTRIDE + 2 * 1024 + 4;  // (r=1,c=0), skew
        const float* x1 = cur + b * BSTRIDE + 3 * 1024 + 4;  // (r=1,c=1), skew
        float acc = 0.f;
        if (t < 1023) acc += x0[t] * P00 + x1[t] * P10;
        if (t >= 1)   acc += x0[t - 1] * P01 + x1[t - 1] * P11;
        A.out[(size_t)(b0 + b) * 1024 + t] = acc;
    }
}

extern "C" void kernel_launch(void* const* d_in, const int* in_sizes, int n_in,
                              void* d_out, int out_size, void* d_ws, size_t ws_size,
                              hipStream_t stream) {
    (void)in_sizes; (void)n_in; (void)d_ws; (void)ws_size; (void)out_size;
    PolyArgs A;
    A.in = (const float*)d_in[0];
    for (int j = 0; j < 10; ++j) {
        A.d1[j] = (const float*)d_in[1 + j];
        A.d2[j] = (const float*)d_in[11 + j];
    }
    A.P   = (const float*)d_in[21];
    A.out = (float*)d_out;
    hipLaunchKernelGGL(hstack_diag_product_kernel,
                       dim3(4096 / B_TILE), dim3(NTHREADS), 0, stream, A);
}